// GeneralLinear_841813590622
// MI455X (gfx1250) — compile-verified
//
#include <hip/hip_runtime.h>

typedef __attribute__((ext_vector_type(2))) float v2f;
typedef __attribute__((ext_vector_type(8))) float v8f;

// ---------------- CDNA5 async / wait helpers (device pass only) -------------
#if defined(__AMDGCN__)
__device__ __forceinline__ void async_copy_b128(unsigned lds_byte, unsigned gl_byte,
                                                const float* base) {
  // GVS mode: per-lane DMA of 16 bytes from global base plus offset gl_byte to LDS address lds_byte; bumps ASYNCcnt
  asm volatile("global_load_async_to_lds_b128 %0, %1, %2"
               :
               : "v"(lds_byte), "v"(gl_byte), "s"(base)
               : "memory");
}
template <int N>
__device__ __forceinline__ void wait_async() {
  asm volatile("s_wait_asynccnt %0" ::"n"(N) : "memory");
}
__device__ __forceinline__ void wait_ds0() {
  asm volatile("s_wait_dscnt 0x0" ::: "memory");
}
#else
__device__ __forceinline__ void async_copy_b128(unsigned, unsigned, const float*) {}
template <int N>
__device__ __forceinline__ void wait_async() {}
__device__ __forceinline__ void wait_ds0() {}
#endif

// Stage one 16-row x chunk (32 k-values = 32*D floats per row) into LDS.
// Per lane: 4*D async b128 transfers. RP = padded row stride (dwords).
template <int D>
__device__ __forceinline__ void stage_chunk(const float* __restrict__ x,
                                            unsigned gbase, unsigned lbase, int lane) {
  constexpr int RP = 32 * D + 4;
#pragma unroll
  for (int t = 0; t < 4 * D; ++t) {
    const int i   = lane + t * 32;       // float4 index within chunk [0, 128*D)
    const int row = i / (8 * D);         // 8*D float4 per row
    const int col = i - row * (8 * D);
    async_copy_b128(lbase + (unsigned)((row * RP + col * 4) * 4),
                    gbase + (unsigned)((row * 9216 + col * 4) * 4), x);
  }
}

// One wave owns 16 rows (n values); computes a 16(n) x 16(o) x D tile with
// v_wmma_f32_16x16x4_f32, K = 1024 in 32 chunks of 32, double-buffered
// async-to-LDS staging, zero in-loop barriers.
template <int D, int OFF, int JOFF>
__device__ __forceinline__ void block_impl(const float* __restrict__ x,
                                           const float* __restrict__ w,
                                           float* __restrict__ out,
                                           float* __restrict__ smem) {
  constexpr int RP   = 32 * D + 4;  // LDS x-row stride (dwords), %64 == 36
  constexpr int NASY = 4 * D;       // async ops per lane per chunk

  float* wl     = smem;             // [1024][16] weights, natural layout
  float* xw_all = smem + 16384;     // per-wave double buffers

  const int tid  = threadIdx.x;
  const int lane = tid & 31;
  const int wave = tid >> 5;
  const int half = lane >> 4;       // 0: K pair (k0,k0+1)  1: (k0+2,k0+3)
  const int Lm   = lane & 15;
  const int n0   = blockIdx.x * 64 + wave * 16;  // this wave's first row

  // One-time: DMA the full 64KB weight matrix into LDS (4096 float4).
  {
    const unsigned wlds = (unsigned)(unsigned long long)wl;
#pragma unroll
    for (int t = 0; t < 32; ++t) {
      const int i = tid + t * 128;
      async_copy_b128(wlds + (unsigned)i * 16u, (unsigned)i * 16u, w);
    }
    wait_async<0>();
  }
  __syncthreads();  // only barrier in the kernel

  float* xw            = xw_all + wave * (2 * 16 * RP);
  const unsigned xlds  = (unsigned)(unsigned long long)xw;
  const unsigned gbase = (unsigned)(n0 * 9216 + OFF) * 4u;

  v8f acc[D];
#pragma unroll
  for (int j = 0; j < D; ++j) acc[j] = {};

  stage_chunk<D>(x, gbase, xlds, lane);  // prologue: chunk 0 -> buffer 0

  for (int cc = 0; cc < 32; ++cc) {
    const int buf = cc & 1;
    if (cc + 1 < 32) {
      wait_ds0();  // drain chunk cc-1 LDS reads before overwriting its buffer
      const int nc        = cc + 1;
      const unsigned goff = (unsigned)(((nc >> 1) * 576 + (nc & 1) * 32 * D) * 4);
      stage_chunk<D>(x, gbase + goff, xlds + (unsigned)((buf ^ 1) * 16 * RP * 4), lane);
      wait_async<NASY>();  // chunk cc complete; chunk cc+1 still in flight
    } else {
      wait_async<0>();
    }

    // A: row M=Lm, K pair (k0+2h, k0+2h+1) -> elements D apart in LDS row.
    const float* xrb = xw + buf * (16 * RP) + Lm * RP + 2 * half * D;
    const int kbase  = cc * 32 + 2 * half;
#pragma unroll
    for (int u0 = 0; u0 < 32; u0 += 4) {
      // B: col N=Lm, pair 16 dwords apart -> ds_load_2addr_b32 offset1:16
      const float* wb = wl + (kbase + u0) * 16 + Lm;
      v2f b;
      b.x = wb[0];
      b.y = wb[16];
#pragma unroll
      for (int j = 0; j < D; ++j) {
        v2f a;
        a.x = xrb[u0 * D + j];
        a.y = xrb[u0 * D + D + j];
#if defined(__gfx1250__)
        acc[j] = __builtin_amdgcn_wmma_f32_16x16x4_f32(
            false, a, false, b, (short)0, acc[j], false, false);
#else
        acc[j][0] += a.x * b.x;  // host-pass placeholder; never executed
#endif
      }
    }
  }

  // D layout: VGPR r, lanes<16 -> M=r, lanes>=16 -> M=r+8; N = Lm.
  // Fold the 1/sqrt(1024) path normalization into the store.
  float* ob = out + (size_t)(n0 + half * 8) * 144 + Lm * 9 + JOFF;
#pragma unroll
  for (int j = 0; j < D; ++j)
#pragma unroll
    for (int r = 0; r < 8; ++r)
      ob[r * 144 + j] = acc[j][r] * 0.03125f;
}

__global__ __launch_bounds__(128) void irrep_linear_fused(
    const float* __restrict__ x, const float* __restrict__ w0,
    const float* __restrict__ w1, const float* __restrict__ w2,
    float* __restrict__ out) {
  extern __shared__ float smem[];
  if (blockIdx.y == 0)
    block_impl<1, 0, 0>(x, w0, out, smem);
  else if (blockIdx.y == 1)
    block_impl<3, 64, 1>(x, w1, out, smem);
  else
    block_impl<5, 256, 4>(x, w2, out, smem);
}

extern "C" void kernel_launch(void* const* d_in, const int* in_sizes, int n_in,
                              void* d_out, int out_size, void* d_ws,
                              size_t ws_size, hipStream_t stream) {
  const float* x  = (const float*)d_in[0];
  const float* w0 = (const float*)d_in[1];
  const float* w1 = (const float*)d_in[2];
  const float* w2 = (const float*)d_in[3];
  float* out      = (float*)d_out;

  // weights 64KB + 4 waves * 2 buffers * 16 rows * (32*5+4) dwords
  const size_t shmem = (size_t)(16384 + 4 * 2 * 16 * (32 * 5 + 4)) * sizeof(float);
  (void)hipFuncSetAttribute((const void*)irrep_linear_fused,
                            hipFuncAttributeMaxDynamicSharedMemorySize, (int)shmem);

  irrep_linear_fused<<<dim3(8192 / 64, 3), dim3(128), shmem, stream>>>(x, w0, w1, w2, out);
}